// NonLocalBlock_58523224375581
// MI455X (gfx1250) — compile-verified
//
#include <hip/hip_runtime.h>
#include <hip/hip_bf16.h>

// NonLocalBlock (self-attention + BN + residual), B=4, N=64*64=4096, C=64, D=32.
// Strategy: fp32 WMMA (v_wmma_f32_16x16x4_f32) everywhere, flash-style fused
// attention (no NxN score materialization), deterministic BN reduction.

typedef float v2f __attribute__((ext_vector_type(2)));
typedef float v8f __attribute__((ext_vector_type(8)));

#define NB    4
#define NPIX  4096            // 64*64
#define NTOT  16384           // NB*NPIX
#define CH    64
#define DIN   32
#define EPSBN 1e-3f

// ---------------------------------------------------------------------------
// Kernel 1: projections  theta/g = X @ W + b (row-major [NTOT,32]),
//           phi stored transposed [B][32][4096] for coalesced B-operand loads.
// One wave (32 threads) per 16-row tile per projection.
// ---------------------------------------------------------------------------
__global__ __launch_bounds__(32) void proj_kernel(
    const float* __restrict__ x,                       // [NTOT, 64]
    const float* __restrict__ gW, const float* __restrict__ gb,
    const float* __restrict__ pW, const float* __restrict__ pb,
    const float* __restrict__ tW, const float* __restrict__ tb,
    float* __restrict__ theta, float* __restrict__ gbuf,
    float* __restrict__ phiT)
{
  const int lane = threadIdx.x;
  const int m    = lane & 15;
  const int h    = lane >> 4;
  const int proj = blockIdx.y;          // 0=g, 1=phi, 2=theta
  const int r0   = blockIdx.x * 16;     // global row

  const float* Wm   = (proj == 0) ? gW : (proj == 1) ? pW : tW;
  const float* bias = (proj == 0) ? gb : (proj == 1) ? pb : tb;

  // A fragments: 16 k-steps over K=64.  A[m][4kk+2h+j]
  v2f a[16];
  const float* xrow = x + (size_t)(r0 + m) * CH;
#pragma unroll
  for (int kk = 0; kk < 16; ++kk) {
    a[kk].x = xrow[4 * kk + 2 * h];
    a[kk].y = xrow[4 * kk + 2 * h + 1];
  }

#pragma unroll
  for (int t = 0; t < 2; ++t) {         // two 16-col n-tiles (D=32)
    v8f acc = {};
#pragma unroll
    for (int kk = 0; kk < 16; ++kk) {
      v2f bf;                           // B[4kk+2h+j][16t+m]
      bf.x = Wm[(4 * kk + 2 * h) * DIN + 16 * t + m];
      bf.y = Wm[(4 * kk + 2 * h + 1) * DIN + 16 * t + m];
      acc  = __builtin_amdgcn_wmma_f32_16x16x4_f32(false, a[kk], false, bf,
                                                   (short)0, acc, false, false);
    }
    const float bv = bias[16 * t + m];
    if (proj == 1) {                    // phi: transposed store [b][d][n]
      const int b  = r0 >> 12;
      const int nb = r0 & (NPIX - 1);
#pragma unroll
      for (int r = 0; r < 8; ++r) {
        const int M = r + 8 * h;
        phiT[((size_t)(b * DIN + 16 * t + m)) * NPIX + nb + M] = acc[r] + bv;
      }
    } else {
      float* outp = (proj == 0) ? gbuf : theta;
#pragma unroll
      for (int r = 0; r < 8; ++r) {
        const int M = r + 8 * h;
        outp[(size_t)(r0 + M) * DIN + 16 * t + m] = acc[r] + bv;
      }
    }
  }
}

// ---------------------------------------------------------------------------
// Kernel 2: fused flash attention + output projection + BN partial stats.
// One wave per 16-row tile; loops over 256 column tiles of its batch.
// Row sums of P are produced by a WMMA against a ones B-matrix so they land
// directly in the accumulator layout (saves ~64 shuffle ops per iteration).
// ---------------------------------------------------------------------------
__global__ __launch_bounds__(32) void attn_kernel(
    const float* __restrict__ theta,    // [NTOT, 32]
    const float* __restrict__ phiT,     // [B][32][4096]
    const float* __restrict__ gbuf,     // [NTOT, 32]
    const float* __restrict__ wW,       // [32, 64]
    const float* __restrict__ wb,       // [64]
    float* __restrict__ wy,             // d_out used as scratch [NTOT, 64]
    float* __restrict__ part1,          // [1024][64] per-wave channel sums
    float* __restrict__ part2)          // [1024][64] per-wave channel sumsq
{
  __shared__ float lds[16 * 36];        // padded (stride 36 -> conflict-free)

  const int lane = threadIdx.x;
  const int m    = lane & 15;
  const int h    = lane >> 4;
  const int rt   = blockIdx.x;          // 0..1023
  const int r0   = rt * 16;             // global row
  const int b    = r0 >> 12;
  const float* phB = phiT + (size_t)b * DIN * NPIX;
  const float* gB  = gbuf + (size_t)(b << 12) * DIN;

  // theta A-fragments (K=32 -> 8 k-steps), loaded once per row tile
  v2f at[8];
  {
    const float* trow = theta + (size_t)(r0 + m) * DIN;
#pragma unroll
    for (int kk = 0; kk < 8; ++kk) {
      at[kk].x = trow[4 * kk + 2 * h];
      at[kk].y = trow[4 * kk + 2 * h + 1];
    }
  }

  float mrun[8];
  v8f y0 = {}, y1 = {}, lsum = {};      // lsum: P @ ones == running row sums
#pragma unroll
  for (int r = 0; r < 8; ++r) mrun[r] = -3.0e38f;

  const v2f ones = {1.0f, 1.0f};

  for (int c0 = 0; c0 < NPIX; c0 += 16) {
    // ---- S = theta @ phi^T (16x16 tile) ----
    v8f S = {};
#pragma unroll
    for (int kk = 0; kk < 8; ++kk) {
      v2f bf;                           // B[k][n] = phi[c0+n][k] (transposed)
      bf.x = phB[(size_t)(4 * kk + 2 * h) * NPIX + c0 + m];
      bf.y = phB[(size_t)(4 * kk + 2 * h + 1) * NPIX + c0 + m];
      S = __builtin_amdgcn_wmma_f32_16x16x4_f32(false, at[kk], false, bf,
                                                (short)0, S, false, false);
    }
    // ---- online softmax: running max + exp ----
    float alpha[8];
#pragma unroll
    for (int r = 0; r < 8; ++r) {
      float rm = S[r];
#pragma unroll
      for (int off = 1; off < 16; off <<= 1)
        rm = fmaxf(rm, __shfl_xor(rm, off));     // stays within 16-lane half
      const float nm = fmaxf(mrun[r], rm);
      alpha[r] = __expf(mrun[r] - nm);
      mrun[r]  = nm;
      S[r]     = __expf(S[r] - nm);              // P tile
    }
#pragma unroll
    for (int r = 0; r < 8; ++r) {
      y0[r]   *= alpha[r];
      y1[r]   *= alpha[r];
      lsum[r] *= alpha[r];
    }
    // ---- stage P: D-layout -> A-layout via LDS ----
    __syncthreads();
#pragma unroll
    for (int r = 0; r < 8; ++r)
      lds[(r + 8 * h) * 36 + m] = S[r];
    __syncthreads();
    v2f pf[4];
#pragma unroll
    for (int kk = 0; kk < 4; ++kk) {
      pf[kk].x = lds[m * 36 + 4 * kk + 2 * h];
      pf[kk].y = lds[m * 36 + 4 * kk + 2 * h + 1];
    }
    // ---- Y += P @ G ; lsum += P @ ones ----
#pragma unroll
    for (int kk = 0; kk < 4; ++kk) {
      const float* grow0 = gB + (size_t)(c0 + 4 * kk + 2 * h) * DIN;
      const float* grow1 = grow0 + DIN;
      v2f b0, b1;
      b0.x = grow0[m];      b0.y = grow1[m];
      b1.x = grow0[16 + m]; b1.y = grow1[16 + m];
      y0 = __builtin_amdgcn_wmma_f32_16x16x4_f32(false, pf[kk], false, b0,
                                                 (short)0, y0, false, false);
      y1 = __builtin_amdgcn_wmma_f32_16x16x4_f32(false, pf[kk], false, b1,
                                                 (short)0, y1, false, false);
      lsum = __builtin_amdgcn_wmma_f32_16x16x4_f32(false, pf[kk], false, ones,
                                                   (short)0, lsum, false, false);
    }
  }

  // ---- normalize by softmax denominator ----
#pragma unroll
  for (int r = 0; r < 8; ++r) {
    const float inv = 1.0f / lsum[r];
    y0[r] *= inv;
    y1[r] *= inv;
  }

  // ---- stage Y (16x32) to LDS for D->A layout ----
  __syncthreads();
#pragma unroll
  for (int r = 0; r < 8; ++r) {
    lds[(r + 8 * h) * 36 + m]      = y0[r];
    lds[(r + 8 * h) * 36 + 16 + m] = y1[r];
  }
  __syncthreads();
  v2f yf[8];
#pragma unroll
  for (int kk = 0; kk < 8; ++kk) {
    yf[kk].x = lds[m * 36 + 4 * kk + 2 * h];
    yf[kk].y = lds[m * 36 + 4 * kk + 2 * h + 1];
  }

  // ---- w_y = Y @ wW + wb ; write + per-channel partial stats ----
#pragma unroll
  for (int tt = 0; tt < 4; ++tt) {
    v8f acc = {};
#pragma unroll
    for (int kk = 0; kk < 8; ++kk) {
      v2f bf;
      bf.x = wW[(4 * kk + 2 * h) * CH + 16 * tt + m];
      bf.y = wW[(4 * kk + 2 * h + 1) * CH + 16 * tt + m];
      acc  = __builtin_amdgcn_wmma_f32_16x16x4_f32(false, yf[kk], false, bf,
                                                   (short)0, acc, false, false);
    }
    const float bv = wb[16 * tt + m];
    float s1 = 0.0f, s2 = 0.0f;
#pragma unroll
    for (int r = 0; r < 8; ++r) {
      const float v = acc[r] + bv;
      s1 += v;
      s2 += v * v;
      wy[(size_t)(r0 + r + 8 * h) * CH + 16 * tt + m] = v;
    }
    s1 += __shfl_xor(s1, 16);           // combine rows 0-7 with 8-15 (same ch)
    s2 += __shfl_xor(s2, 16);
    if (lane < 16) {
      part1[rt * CH + 16 * tt + lane] = s1;
      part2[rt * CH + 16 * tt + lane] = s2;
    }
  }
}

// ---------------------------------------------------------------------------
// Kernel 3: deterministic BN stat reduction -> per-channel scale/shift
// ---------------------------------------------------------------------------
__global__ __launch_bounds__(64) void bnstats_kernel(
    const float* __restrict__ part1, const float* __restrict__ part2,
    const float* __restrict__ gamma, const float* __restrict__ beta,
    float* __restrict__ scale, float* __restrict__ shift)
{
  const int c = threadIdx.x;
  if (c < CH) {
    float s = 0.0f, q = 0.0f;
    for (int w = 0; w < 1024; ++w) {
      s += part1[w * CH + c];
      q += part2[w * CH + c];
    }
    const float invN = 1.0f / (float)NTOT;
    const float mean = s * invN;
    const float var  = q * invN - mean * mean;
    const float sc   = gamma[c] * rsqrtf(var + EPSBN);
    scale[c] = sc;
    shift[c] = beta[c] - mean * sc;
  }
}

// ---------------------------------------------------------------------------
// Kernel 4: out = x + w_y * scale[c] + shift[c]   (float4 vectorized)
// ---------------------------------------------------------------------------
__global__ __launch_bounds__(256) void final_kernel(
    const float* __restrict__ x, const float* __restrict__ scale,
    const float* __restrict__ shift, float* __restrict__ out)
{
  const int i = blockIdx.x * blockDim.x + threadIdx.x;   // float4 index
  const float4* x4 = (const float4*)x;
  float4*       o4 = (float4*)out;
  const float4 wv = o4[i];
  const float4 xv = x4[i];
  const int c = (i & 15) * 4;                            // (i*4) % 64
  float4 r;
  r.x = xv.x + wv.x * scale[c + 0] + shift[c + 0];
  r.y = xv.y + wv.y * scale[c + 1] + shift[c + 1];
  r.z = xv.z + wv.z * scale[c + 2] + shift[c + 2];
  r.w = xv.w + wv.w * scale[c + 3] + shift[c + 3];
  o4[i] = r;
}

// ---------------------------------------------------------------------------
extern "C" void kernel_launch(void* const* d_in, const int* in_sizes, int n_in,
                              void* d_out, int out_size, void* d_ws, size_t ws_size,
                              hipStream_t stream) {
  const float* x     = (const float*)d_in[0];
  const float* gW    = (const float*)d_in[1];
  const float* gb    = (const float*)d_in[2];
  const float* pW    = (const float*)d_in[3];
  const float* pb    = (const float*)d_in[4];
  const float* tW    = (const float*)d_in[5];
  const float* tb    = (const float*)d_in[6];
  const float* wW    = (const float*)d_in[7];
  const float* wb    = (const float*)d_in[8];
  const float* gamma = (const float*)d_in[9];
  const float* beta  = (const float*)d_in[10];
  float* out = (float*)d_out;

  // Workspace layout (floats): ~6.8 MB total
  float* ws    = (float*)d_ws;
  float* theta = ws;                         // 16384*32
  float* gbuf  = ws + 524288;                // 16384*32
  float* phiT  = ws + 1048576;               // 4*32*4096
  float* part1 = ws + 1572864;               // 1024*64
  float* part2 = ws + 1638400;               // 1024*64
  float* scale = ws + 1703936;               // 64
  float* shift = ws + 1704000;               // 64

  proj_kernel<<<dim3(NTOT / 16, 3), 32, 0, stream>>>(
      x, gW, gb, pW, pb, tW, tb, theta, gbuf, phiT);

  attn_kernel<<<NTOT / 16, 32, 0, stream>>>(
      theta, phiT, gbuf, wW, wb, out, part1, part2);

  bnstats_kernel<<<1, 64, 0, stream>>>(part1, part2, gamma, beta, scale, shift);

  final_kernel<<<(NTOT * CH / 4) / 256, 256, 0, stream>>>(x, scale, shift, out);
}